// MaskedHeteroGAT_45835890983027
// MI455X (gfx1250) — compile-verified
//
#include <hip/hip_runtime.h>
#include <hip/hip_bf16.h>
#include <math.h>

typedef __attribute__((ext_vector_type(2))) float v2f;
typedef __attribute__((ext_vector_type(8))) float v8f;

#define GAT_H 4
#define GAT_C 64
#define GAT_HC 256
#define GAT_F 128
#define NEG_SLOPE 0.2f

// ---------- order-preserving float<->uint for atomic max on signed floats ----------
__device__ __forceinline__ unsigned enc_ord(float f) {
  unsigned u = __float_as_uint(f);
  return (u & 0x80000000u) ? ~u : (u | 0x80000000u);
}
__device__ __forceinline__ float dec_ord(unsigned u) {
  return (u & 0x80000000u) ? __uint_as_float(u & 0x7FFFFFFFu)
                           : __uint_as_float(~u);
}

// ---------- utility ----------
__global__ void k_zero(float* __restrict__ p, int n) {
  int i = blockIdx.x * blockDim.x + threadIdx.x;
  if (i < n) p[i] = 0.0f;
}

__global__ void k_relu(float* __restrict__ p, int n) {
  int i = blockIdx.x * blockDim.x + threadIdx.x;
  if (i < n) { float v = p[i]; p[i] = v > 0.0f ? v : 0.0f; }
}

// ---------- fp32 WMMA GEMM: Y[M,N] = (A * rowmask) @ B ----------
// A row-major [M,K], B row-major [K,N].
// One wave computes a 16x64 strip of Y: 4 independent 16x16 accumulators, so
// each A fragment is reused 4x per K-step and the 4 WMMA chains overlap.
// V_WMMA_F32_16X16X4_F32 operand layouts per CDNA5 ISA 7.12.2:
//   A 16x4:  VGPR0 = K{0,2} (lanes 0-15 / 16-31), VGPR1 = K{1,3}
//   B 4x16:  VGPR0 = rows K{0,2}, VGPR1 = rows K{1,3}, N striped over lanes 0-15
//   C/D:     VGPR v = rows {v, v+8} selected by lane half
__global__ void __launch_bounds__(256)
k_gemm_wmma_f32(const float* __restrict__ A,
                const float* __restrict__ rowmask,
                const float* __restrict__ B,
                float* __restrict__ Y,
                int M, int N, int K) {
  const int wave   = (blockIdx.x * blockDim.x + threadIdx.x) >> 5;
  const int lane   = threadIdx.x & 31;
  const int groupsN = N >> 6;                // 64-column groups
  const int tm     = wave / groupsN;
  const int tg     = wave - tm * groupsN;
  if (tm * 16 >= M) return;                  // uniform per wave (exact tiling here)

  const int half = lane >> 4;                // 0: lanes 0-15, 1: lanes 16-31
  const int lm   = lane & 15;
  const int am   = tm * 16 + lm;             // A row this lane feeds
  const float msk = rowmask ? rowmask[am] : 1.0f;
  const float* __restrict__ Arow  = A + (size_t)am * K;
  const float* __restrict__ Bbase = B + tg * 64 + lm;

  v8f acc0 = {0.f,0.f,0.f,0.f,0.f,0.f,0.f,0.f};
  v8f acc1 = acc0, acc2 = acc0, acc3 = acc0;

  for (int k0 = 0; k0 < K; k0 += 4) {
    const int ka = k0 + half * 2;
    v2f a;
    a.x = Arow[ka]     * msk;
    a.y = Arow[ka + 1] * msk;
    const float* __restrict__ br0 = Bbase + (size_t)ka * N;        // B row ka
    const float* __restrict__ br1 = Bbase + (size_t)(ka + 1) * N;  // B row ka+1
    v2f b;
    b.x = br0[0];  b.y = br1[0];
    acc0 = __builtin_amdgcn_wmma_f32_16x16x4_f32(false, a, false, b, (short)0, acc0, false, false);
    b.x = br0[16]; b.y = br1[16];
    acc1 = __builtin_amdgcn_wmma_f32_16x16x4_f32(false, a, false, b, (short)0, acc1, false, false);
    b.x = br0[32]; b.y = br1[32];
    acc2 = __builtin_amdgcn_wmma_f32_16x16x4_f32(false, a, false, b, (short)0, acc2, false, false);
    b.x = br0[48]; b.y = br1[48];
    acc3 = __builtin_amdgcn_wmma_f32_16x16x4_f32(false, a, false, b, (short)0, acc3, false, false);
  }

  const int mb = tm * 16 + half * 8;
  float* __restrict__ Yb = Y + (size_t)mb * N + tg * 64 + lm;
#pragma unroll
  for (int v = 0; v < 8; ++v) {
    float* __restrict__ yr = Yb + (size_t)v * N;
    yr[0]  = acc0[v];
    yr[16] = acc1[v];
    yr[32] = acc2[v];
    yr[48] = acc3[v];
  }
}

// ---------- edge scoring: e[e,h] = a[h,:] . leaky_relu(hs[src]+hd[dst]); segment max ----------
__global__ void __launch_bounds__(256)
k_edge_score(const float* __restrict__ hs, const float* __restrict__ hd,
             const int* __restrict__ src, const int* __restrict__ dst,
             const float* __restrict__ att,        // [256] = a[h*64+c]
             float* __restrict__ ebuf,             // [E*4]
             unsigned* __restrict__ emax,          // [Nt*4] order-encoded
             int E) {
  const int e = (blockIdx.x * blockDim.x + threadIdx.x) >> 5;
  if (e >= E) return;
  const int lane = threadIdx.x & 31;
  const int s = src[e], d = dst[e];
  const float* __restrict__ ps = hs + (size_t)s * GAT_HC;
  const float* __restrict__ pd = hd + (size_t)d * GAT_HC;
#pragma unroll
  for (int h = 0; h < GAT_H; ++h) {
    float p = 0.0f;
#pragma unroll
    for (int i = 0; i < 2; ++i) {              // C=64 -> 2 elems/lane
      const int c = h * GAT_C + lane + i * 32;
      float z = ps[c] + pd[c];
      z = (z > 0.0f) ? z : NEG_SLOPE * z;
      p += att[c] * z;
    }
#pragma unroll
    for (int off = 16; off >= 1; off >>= 1)
      p += __shfl_xor(p, off, 32);
    if (lane == 0) {
      ebuf[e * GAT_H + h] = p;
      atomicMax(&emax[d * GAT_H + h], enc_ord(p));
    }
  }
}

// ---------- softmax numerator + segment denom ----------
__global__ void k_edge_w(const float* __restrict__ ebuf,
                         const unsigned* __restrict__ emax,
                         const int* __restrict__ dst,
                         const float* __restrict__ emask,
                         float* __restrict__ wbuf,
                         float* __restrict__ denom,
                         int E) {
  const int idx = blockIdx.x * blockDim.x + threadIdx.x;
  if (idx >= E * GAT_H) return;
  const int e = idx >> 2, h = idx & 3;
  const int d = dst[e];
  const float m = dec_ord(emax[d * GAT_H + h]);
  const float w = expf(ebuf[idx] - m) * emask[e];
  wbuf[idx] = w;
  atomicAdd(&denom[d * GAT_H + h], w);
}

// ---------- weighted scatter-add aggregation: out[dst] += alpha * hs[src] ----------
__global__ void __launch_bounds__(256)
k_edge_aggr(const float* __restrict__ wbuf, const float* __restrict__ denom,
            const int* __restrict__ src, const int* __restrict__ dst,
            const float* __restrict__ hs, float* __restrict__ out, int E) {
  const int e = (blockIdx.x * blockDim.x + threadIdx.x) >> 5;
  if (e >= E) return;
  const int lane = threadIdx.x & 31;
  const int s = src[e], d = dst[e];
  float coeff[GAT_H];
#pragma unroll
  for (int h = 0; h < GAT_H; ++h)
    coeff[h] = wbuf[e * GAT_H + h] / (denom[d * GAT_H + h] + 1e-16f);
  const float* __restrict__ ps = hs + (size_t)s * GAT_HC;
  float* __restrict__ po = out + (size_t)d * GAT_HC;
#pragma unroll
  for (int i = 0; i < 8; ++i) {               // 256 channels / 32 lanes
    const int c = lane + i * 32;              // head = i>>1 (uniform across lanes)
    atomicAdd(&po[c], coeff[i >> 1] * ps[c]);
  }
}

// ---------- column mean of h2 (partial per block, atomicAdd) ----------
__global__ void k_pool(const float* __restrict__ h2, float* __restrict__ pooled, int rows) {
  const int c  = threadIdx.x;                 // 256 channels
  const int r0 = blockIdx.x * 64;
  int r1 = r0 + 64; if (r1 > rows) r1 = rows;
  float s = 0.0f;
  for (int r = r0; r < r1; ++r) s += h2[(size_t)r * GAT_HC + c];
  atomicAdd(&pooled[c], s);
}

// ---------- per-type edge_attr mean ----------
__global__ void k_agg_edge(const float* __restrict__ edge_attr, float* __restrict__ agg, int E) {
  __shared__ float red[256];
  const float* __restrict__ p = edge_attr + (size_t)blockIdx.x * E;
  float s = 0.0f;
  for (int i = threadIdx.x; i < E; i += blockDim.x) s += p[i];
  red[threadIdx.x] = s;
  __syncthreads();
  for (int off = 128; off >= 1; off >>= 1) {
    if (threadIdx.x < off) red[threadIdx.x] += red[threadIdx.x + off];
    __syncthreads();
  }
  if (threadIdx.x == 0) agg[blockIdx.x] = red[0] / (float)E;
}

// ---------- classifier: sigmoid(feat @ W_cls + b) ----------
__global__ void k_final(const float* __restrict__ pooled, float invRows,
                        const float* __restrict__ agg,
                        const float* __restrict__ Wc, const float* __restrict__ bc,
                        float* __restrict__ out, int T) {
  __shared__ float red[256];
  float s = pooled[threadIdx.x] * invRows * Wc[threadIdx.x];
  if ((int)threadIdx.x < T) s += agg[threadIdx.x] * Wc[GAT_HC + threadIdx.x];
  red[threadIdx.x] = s;
  __syncthreads();
  for (int off = 128; off >= 1; off >>= 1) {
    if (threadIdx.x < off) red[threadIdx.x] += red[threadIdx.x + off];
    __syncthreads();
  }
  if (threadIdx.x == 0) {
    const float logit = red[0] + bc[0];
    out[0] = 1.0f / (1.0f + expf(-logit));
  }
}

// =====================================================================
extern "C" void kernel_launch(void* const* d_in, const int* in_sizes, int n_in,
                              void* d_out, int out_size, void* d_ws, size_t ws_size,
                              hipStream_t stream) {
  const float* x_src     = (const float*)d_in[0];   // [Ns,F]
  const float* x_tgt     = (const float*)d_in[1];   // [T,Nt,F]
  const float* edge_attr = (const float*)d_in[2];   // [T,E,1]
  const float* node_mask = (const float*)d_in[3];   // [Ns]
  const float* edge_mask = (const float*)d_in[4];   // [E]
  const int*   edge_idx  = (const int*)  d_in[5];   // [T,2,E]
  const float* W1_src    = (const float*)d_in[6];   // [T,F,HC]
  const float* W1_dst    = (const float*)d_in[7];   // [T,F,HC]
  const float* a1        = (const float*)d_in[8];   // [T,H,C]
  const float* W2_src    = (const float*)d_in[9];   // [T,F,HC]
  const float* W2_dst    = (const float*)d_in[10];  // [T,HC,HC]
  const float* a2        = (const float*)d_in[11];  // [T,H,C]
  const float* W_cls     = (const float*)d_in[12];  // [HC+T,1]
  const float* b_cls     = (const float*)d_in[13];  // [1]
  float* out = (float*)d_out;

  const int Ns = in_sizes[0] / GAT_F;
  const int T  = in_sizes[8] / GAT_HC;
  const int E  = in_sizes[4];
  const int Nt = in_sizes[1] / (T * GAT_F);
  const int t  = T - 1;                     // only the last type reaches the output

  // ---- workspace carve-up (floats) ----
  float* ws     = (float*)d_ws;
  float* p_hs   = ws;                                   // [Ns,256]
  float* p_hd   = p_hs  + (size_t)Ns * GAT_HC;          // [Nt,256]
  float* p_h1   = p_hd  + (size_t)Nt * GAT_HC;          // [Nt,256]
  float* p_h2   = p_h1  + (size_t)Nt * GAT_HC;          // [Nt,256]
  float* p_e    = p_h2  + (size_t)Nt * GAT_HC;          // [E,4]
  float* p_w    = p_e   + (size_t)E * GAT_H;            // [E,4]
  float* p_den  = p_w   + (size_t)E * GAT_H;            // [Nt,4]
  unsigned* p_emax = (unsigned*)(p_den + (size_t)Nt * GAT_H); // [Nt,4]
  float* p_pool = (float*)(p_emax + (size_t)Nt * GAT_H);      // [256]
  float* p_agg  = p_pool + GAT_HC;                            // [T]

  const int* src = edge_idx + (size_t)t * 2 * E;
  const int* dst = src + E;

  const int ZB = 256;
  auto zgrid = [](long n) { return (int)((n + 255) / 256); };
  // one wave per 16x64 strip
  const int gemm_waves_S  = (Ns / 16) * (GAT_HC / 64);
  const int gemm_blocks_S = (gemm_waves_S * 32 + 255) / 256;
  const int gemm_waves_T  = (Nt / 16) * (GAT_HC / 64);
  const int gemm_blocks_T = (gemm_waves_T * 32 + 255) / 256;
  const int ewave_blocks  = (E * 32 + 255) / 256;       // 1 wave / edge
  const int ew_blocks     = (E * GAT_H + 255) / 256;

  // ===================== layer 1 =====================
  k_zero<<<zgrid((long)Nt * GAT_HC), ZB, 0, stream>>>(p_h1, Nt * GAT_HC);
  k_zero<<<zgrid((long)2 * Nt * GAT_H), ZB, 0, stream>>>(p_den, 2 * Nt * GAT_H); // denom + emax(=0 bits)

  k_gemm_wmma_f32<<<gemm_blocks_S, 256, 0, stream>>>(
      x_src, node_mask, W1_src + (size_t)t * GAT_F * GAT_HC, p_hs, Ns, GAT_HC, GAT_F);
  k_gemm_wmma_f32<<<gemm_blocks_T, 256, 0, stream>>>(
      x_tgt + (size_t)t * Nt * GAT_F, node_mask,
      W1_dst + (size_t)t * GAT_F * GAT_HC, p_hd, Nt, GAT_HC, GAT_F);

  k_edge_score<<<ewave_blocks, 256, 0, stream>>>(p_hs, p_hd, src, dst,
      a1 + (size_t)t * GAT_HC, p_e, p_emax, E);
  k_edge_w<<<ew_blocks, 256, 0, stream>>>(p_e, p_emax, dst, edge_mask, p_w, p_den, E);
  k_edge_aggr<<<ewave_blocks, 256, 0, stream>>>(p_w, p_den, src, dst, p_hs, p_h1, E);
  k_relu<<<zgrid((long)Nt * GAT_HC), ZB, 0, stream>>>(p_h1, Nt * GAT_HC);

  // ===================== layer 2 =====================
  k_zero<<<zgrid((long)Nt * GAT_HC), ZB, 0, stream>>>(p_h2, Nt * GAT_HC);
  k_zero<<<zgrid((long)2 * Nt * GAT_H), ZB, 0, stream>>>(p_den, 2 * Nt * GAT_H);

  k_gemm_wmma_f32<<<gemm_blocks_S, 256, 0, stream>>>(
      x_src, node_mask, W2_src + (size_t)t * GAT_F * GAT_HC, p_hs, Ns, GAT_HC, GAT_F);
  k_gemm_wmma_f32<<<gemm_blocks_T, 256, 0, stream>>>(
      p_h1, (const float*)nullptr, W2_dst + (size_t)t * GAT_HC * GAT_HC, p_hd, Nt, GAT_HC, GAT_HC);

  k_edge_score<<<ewave_blocks, 256, 0, stream>>>(p_hs, p_hd, src, dst,
      a2 + (size_t)t * GAT_HC, p_e, p_emax, E);
  k_edge_w<<<ew_blocks, 256, 0, stream>>>(p_e, p_emax, dst, edge_mask, p_w, p_den, E);
  k_edge_aggr<<<ewave_blocks, 256, 0, stream>>>(p_w, p_den, src, dst, p_hs, p_h2, E);

  // ===================== readout =====================
  k_zero<<<1, 256, 0, stream>>>(p_pool, GAT_HC);
  k_pool<<<(Nt + 63) / 64, 256, 0, stream>>>(p_h2, p_pool, Nt);
  k_agg_edge<<<T, 256, 0, stream>>>(edge_attr, p_agg, E);
  k_final<<<1, 256, 0, stream>>>(p_pool, 1.0f / (float)Nt, p_agg, W_cls, b_cls, out, T);
}